// ShallowWaterStepper_87540023427704
// MI455X (gfx1250) — compile-verified
//
#include <hip/hip_runtime.h>

// ---------------------------------------------------------------------------
// Shallow-water RK4 step with latitude-banded DISCO convolution as fp32 WMMA
// band-GEMMs (V_WMMA_F32_16X16X4_F32) on gfx1250.
//
//   out(l,x) = sum_{d=-2..2} sum_p K[l,d,p] * in(clip(l+d), (x+p) mod NLON)
//
// Per 16-lat output tile: D[16 lat x 16 lon] += A[16x4] * B[4x16], reduction
// k = (input row l' in the 20-row band, chunks of 4) x (longitude tap p).
//  - A: kernel weights, read branch-free from a zero-padded transposed layout
//       Kt[p][l][8] = {0, K(d=-2..2), 0, 0}; the p-part of the address is
//       wave-uniform, the lane part is loop-invariant.
//  - B: 20 full input rings staged in LDS by the Tensor Data Mover (clamped
//       polar rows fixed up with async global->LDS loads).
//  - XB longitude tiles per wave give XB*NK independent WMMA chains.
// ---------------------------------------------------------------------------

#define NLAT 721
#define NLON 1440
#define NB 4
#define TILE_M 16
#define ROWS 20                 // TILE_M + 2*KOFF
#define NXT (NLON / 16)         // 90 longitude tiles
#define NTILES 46               // ceil(NLAT / 16)
#define NTOT ((size_t)NB * NLAT * NLON)
#define KTSZ ((size_t)NLON * NLAT * 8)   // padded transposed kernel, floats

#define G_CONST 1.0f
#define H0_CONST 1.0f
#define DT_CONST 0.01f

typedef float v2f __attribute__((ext_vector_type(2)));
typedef float v8f __attribute__((ext_vector_type(8)));
typedef unsigned int u32;
typedef u32 v4u __attribute__((ext_vector_type(4)));
typedef int v4i __attribute__((ext_vector_type(4)));
typedef int v8i __attribute__((ext_vector_type(8)));

__device__ __forceinline__ int imin(int a, int b) { return a < b ? a : b; }
__device__ __forceinline__ int imax(int a, int b) { return a > b ? a : b; }

// Async global -> LDS copy of 16 bytes per lane (VGLOBAL async path, ASYNCcnt).
__device__ __forceinline__ void async_copy16(void* lds_dst, const void* gsrc) {
    unsigned lds_off = (unsigned)(unsigned long long)lds_dst;  // low 32 = LDS addr
    unsigned long long ga = (unsigned long long)gsrc;
    asm volatile("global_load_async_to_lds_b128 %0, %1, off"
                 :: "v"(lds_off), "v"(ga) : "memory");
}
__device__ __forceinline__ void wait_async_all() {
    asm volatile("s_wait_asynccnt 0x0" ::: "memory");
}

// ---------------------------------------------------------------------------
// Setup 1: per lat-tile max wrapped tap half-width W from kernel nonzeros.
// ---------------------------------------------------------------------------
__global__ __launch_bounds__(256) void setup_w_kernel(
    const float* __restrict__ Ka, const float* __restrict__ Kb,
    const float* __restrict__ Kc, const float* __restrict__ Kd,
    int* __restrict__ Wt)
{
    __shared__ int wmax;
    if (threadIdx.x == 0) wmax = 1;
    __syncthreads();
    int l0 = blockIdx.x * TILE_M;
    int local = 0;
    for (int rd = 0; rd < TILE_M * 5; ++rd) {
        int l = l0 + rd / 5;
        if (l >= NLAT) break;
        size_t base = ((size_t)l * 5 + (rd % 5)) * NLON;
        for (int p = threadIdx.x; p < NLON; p += blockDim.x) {
            float s = fabsf(Ka[base + p]) + fabsf(Kb[base + p]) +
                      fabsf(Kc[base + p]) + fabsf(Kd[base + p]);
            if (s != 0.0f) {
                int wp = (p <= NLON / 2) ? p : (NLON - p);
                local = imax(local, wp);
            }
        }
    }
    atomicMax(&wmax, local);
    __syncthreads();
    if (threadIdx.x == 0) Wt[blockIdx.x] = wmax;
}

// ---------------------------------------------------------------------------
// Setup 2: K[l,d,p] (721x5x1440) -> Kt[p][l][8] = {0, K(d=-2..2), 0, 0}.
// LDS-tiled so both global sides stay coalesced.
// ---------------------------------------------------------------------------
#define TKL 32
__global__ __launch_bounds__(256) void transpose_pad_kernel(
    const float* __restrict__ K, float* __restrict__ Kt)
{
    __shared__ float tb[TKL * 5 * (TKL + 1)];
    const int pm0 = blockIdx.x * TKL;
    const int l0  = blockIdx.y * TKL;
    for (int e = threadIdx.x; e < TKL * 5 * TKL; e += blockDim.x) {
        int pml = e & (TKL - 1);
        int ld  = e / TKL;                 // l_local*5 + d
        int l   = l0 + ld / 5;
        float v = 0.0f;
        if (l < NLAT) v = K[((size_t)l * 5 + (ld % 5)) * NLON + pm0 + pml];
        tb[ld * (TKL + 1) + pml] = v;
    }
    __syncthreads();
    for (int e = threadIdx.x; e < TKL * TKL; e += blockDim.x) {
        int ll  = e & (TKL - 1);
        int pml = e / TKL;
        int l   = l0 + ll;
        if (l >= NLAT) continue;
        float* dst = Kt + ((size_t)(pm0 + pml) * NLAT + l) * 8;
        dst[0] = 0.0f;
#pragma unroll
        for (int d = 0; d < 5; ++d) dst[1 + d] = tb[(ll * 5 + d) * (TKL + 1) + pml];
        dst[6] = 0.0f;
        dst[7] = 0.0f;
    }
}

// ---------------------------------------------------------------------------
// Banded correlation. One workgroup = (lat tile, batch); 8 waves sweep the
// longitude tiles in blocks of XB. NK = kernels applied to the staged field.
// ---------------------------------------------------------------------------
template <int NK, int XB>
__global__ __launch_bounds__(256) void corr_kernel(
    const float* __restrict__ src,
    const float* __restrict__ K1t, const float* __restrict__ K2t,
    float* __restrict__ out1, float* __restrict__ out2,
    const int* __restrict__ Wt, int accum)
{
    __shared__ float srows[ROWS * NLON];   // 20 full rings, 112.5 KB
    const int tileL = blockIdx.x;
    const int b     = blockIdx.y;
    const int l0    = tileL * TILE_M;
    const int tid   = threadIdx.x;

    // ---- stage 20 clamped rows (TDM for the valid range, async for clamps) ----
    const int jlo = imax(0, 2 - l0);                       // first j with gl >= 0
    const int jhi = imin(ROWS - 1, (NLAT - 1) - (l0 - 2)); // last j with gl <= 720
    const size_t srcb = (size_t)b * NLAT * NLON;
#if __has_builtin(__builtin_amdgcn_tensor_load_to_lds)
    if (tid == 0) {
        const int nrows = jhi - jlo + 1;
        unsigned long long ga =
            (unsigned long long)(src + srcb + (size_t)(l0 - 2 + jlo) * NLON);
        u32 ldsa = (u32)(unsigned long long)&srows[jlo * NLON];
        v4u g0;
        g0[0] = 1u;                                    // count=1 descriptor
        g0[1] = ldsa;                                  // lds_addr
        g0[2] = (u32)(ga & 0xffffffffu);               // global_addr lo
        g0[3] = (u32)((ga >> 32) & 0x1ffffffu) | (2u << 30);  // addr hi | type=2
        v8i g1;
        g1[0] = (int)(2u << 16);                       // data_size = 4B
        g1[1] = (int)((u32)NLON << 16);                // tensor_dim0 lo16
        g1[2] = (int)((u32)nrows << 16);               // dim0 hi | tensor_dim1 lo16
        g1[3] = (int)((u32)NLON << 16);                // dim1 hi | tile_dim0 = 1440
        g1[4] = nrows;                                 // tile_dim1 | tile_dim2=0
        g1[5] = NLON;                                  // tensor_dim0_stride lo
        g1[6] = 0;                                     // stride hi | dim1_stride lo
        g1[7] = 0;
        v4i g2; g2[0] = 1; g2[1] = 1; g2[2] = 0; g2[3] = 0;   // dims 2/3 inert
        v4i g3; g3[0] = 0; g3[1] = (int)(1u << 16); g3[2] = 0; g3[3] = 0;
        v8i g4 = {0, 0, 0, 0, 0, 0, 0, 0};             // unused trailing group
        __builtin_amdgcn_tensor_load_to_lds(g0, g1, g2, g3, g4, 0);
    }
    if (jlo > 0 || jhi < ROWS - 1) {
        for (int c = tid; c < ROWS * (NLON / 4); c += blockDim.x) {
            int j = c / (NLON / 4);
            if (j >= jlo && j <= jhi) continue;
            int xi = (c % (NLON / 4)) * 4;
            int gl = (j < jlo) ? 0 : (NLAT - 1);
            async_copy16(&srows[j * NLON + xi], src + srcb + (size_t)gl * NLON + xi);
        }
    }
    __builtin_amdgcn_s_wait_tensorcnt(0);
#else
    for (int c = tid; c < ROWS * (NLON / 4); c += blockDim.x) {
        int j  = c / (NLON / 4);
        int xi = (c % (NLON / 4)) * 4;
        int gl = imin(imax(l0 - 2 + j, 0), NLAT - 1);
        async_copy16(&srows[j * NLON + xi], src + srcb + (size_t)gl * NLON + xi);
    }
#endif
    wait_async_all();
    __syncthreads();

    const int lane  = tid & 31;
    const int wv    = tid >> 5;
    const int mrow  = lane & 15;     // A: output-lat row; B/D: lon column
    const int khalf = lane >> 4;     // K-half: reg r holds K-index 2*khalf + r
    const int W     = Wt[tileL];
    const int plo   = -W;
    const int phi   = (W == NLON / 2) ? (W - 1) : W;
    const bool mValid = (l0 + mrow) < NLAT;
    const int lA = imin(l0 + mrow, NLAT - 1);

    // Loop-invariant (per-lane) A offsets into the padded row Kt[p][l][0..7].
    int aoff[5];
#pragma unroll
    for (int ch = 0; ch < 5; ++ch) {
        int t = ch * 4 + 2 * khalf - mrow;             // d0 + 2
        int base = (mValid && t >= -1 && t <= 4) ? (t + 1) : 6;  // 6,7 = zeros
        aoff[ch] = lA * 8 + base;
    }

    for (int xt0 = wv * XB; xt0 < NXT; xt0 += 8 * XB) {
        const int xb0 = xt0 * 16;
        v8f acc1[XB];
        v8f acc2[XB];
#pragma unroll
        for (int t = 0; t < XB; ++t) { acc1[t] = (v8f){}; acc2[t] = (v8f){}; }

        for (int p = plo; p <= phi; ++p) {
            const int pm = (p < 0) ? p + NLON : p;
            const float* k1p = K1t + (size_t)pm * (NLAT * 8);
            const float* k2p = K2t + (size_t)pm * (NLAT * 8);
            int xs[XB];
#pragma unroll
            for (int t = 0; t < XB; ++t) {
                int x = xb0 + t * 16 + mrow + p;
                if (x < 0) x += NLON;
                if (x >= NLON) x -= NLON;
                if (x >= NLON) x -= NLON;
                xs[t] = x;
            }
#pragma unroll
            for (int ch = 0; ch < 5; ++ch) {
                v2f a1, a2;
                a1.x = k1p[aoff[ch]];
                a1.y = k1p[aoff[ch] + 1];
                if (NK > 1) { a2.x = k2p[aoff[ch]]; a2.y = k2p[aoff[ch] + 1]; }
                const int k0 = ch * 4 + 2 * khalf;
#pragma unroll
                for (int t = 0; t < XB; ++t) {
                    v2f bb;
                    bb.x = srows[k0 * NLON + xs[t]];
                    bb.y = srows[(k0 + 1) * NLON + xs[t]];
                    acc1[t] = __builtin_amdgcn_wmma_f32_16x16x4_f32(
                        false, a1, false, bb, (short)0, acc1[t], false, false);
                    if (NK > 1)
                        acc2[t] = __builtin_amdgcn_wmma_f32_16x16x4_f32(
                            false, a2, false, bb, (short)0, acc2[t], false, false);
                }
            }
        }
        // D layout: VGPR r -> M = r + 8*khalf, N = lane&15.
#pragma unroll
        for (int t = 0; t < XB; ++t) {
            if (xt0 + t >= NXT) break;
#pragma unroll
            for (int r = 0; r < 8; ++r) {
                int gl = l0 + r + 8 * khalf;
                if (gl < NLAT) {
                    size_t oi = ((size_t)b * NLAT + gl) * NLON + xb0 + t * 16 + mrow;
                    float v1 = acc1[t][r];
                    out1[oi] = accum ? (out1[oi] + v1) : v1;
                    if (NK > 1) out2[oi] = acc2[t][r];
                }
            }
        }
    }
}

// ---------------------------------------------------------------------------
// Pointwise: init state + zero accumulators.
// ---------------------------------------------------------------------------
__global__ void init_kernel(const float* __restrict__ h, const float* __restrict__ uv,
                            float* __restrict__ ch, float* __restrict__ cu,
                            float* __restrict__ cv, float* __restrict__ ah,
                            float* __restrict__ au, float* __restrict__ av)
{
    size_t i = (size_t)blockIdx.x * blockDim.x + threadIdx.x;
    if (i >= NTOT) return;
    ch[i] = h[i];
    cu[i] = uv[2 * i];
    cv[i] = uv[2 * i + 1];
    ah[i] = 0.0f; au[i] = 0.0f; av[i] = 0.0f;
}

// ---------------------------------------------------------------------------
// Pointwise RK stage combine.
// ---------------------------------------------------------------------------
__global__ void update_kernel(
    const float* __restrict__ h0, const float* __restrict__ uv0,
    const float* __restrict__ fcor,
    const float* __restrict__ rawH, const float* __restrict__ rawC,
    const float* __restrict__ rawS,
    float* __restrict__ ch, float* __restrict__ cu, float* __restrict__ cv,
    float* __restrict__ ah, float* __restrict__ au, float* __restrict__ av,
    float* __restrict__ out, float wacc, float cstage, int final_stage)
{
    size_t i = (size_t)blockIdx.x * blockDim.x + threadIdx.x;
    if (i >= NTOT) return;
    int l = (int)((i / NLON) % NLAT);
    float f  = fcor[l];
    float uc = cu[i], vc = cv[i];
    float th = -H0_CONST * rawH[i];
    float tu = -G_CONST * rawC[i] - f * vc;
    float tv = -G_CONST * rawS[i] + f * uc;
    float nah = ah[i] + wacc * th;
    float nau = au[i] + wacc * tu;
    float nav = av[i] + wacc * tv;
    ah[i] = nah; au[i] = nau; av[i] = nav;
    float bh = h0[i], bu = uv0[2 * i], bv = uv0[2 * i + 1];
    if (final_stage) {
        const float k = DT_CONST / 6.0f;
        out[i]                = bh + k * nah;
        out[NTOT + 2 * i]     = bu + k * nau;
        out[NTOT + 2 * i + 1] = bv + k * nav;
    } else {
        ch[i] = bh + cstage * th;
        cu[i] = bu + cstage * tu;
        cv[i] = bv + cstage * tv;
    }
}

// ---------------------------------------------------------------------------
// Host side. Workspace (floats):
//   [0..9)*NTOT : cur_h,cur_u,cur_v, raw_h,raw_c,raw_s, acc_h,acc_u,acc_v
//   then 4 * KTSZ padded transposed kernels, then NTILES ints.  (~283 MB)
// ---------------------------------------------------------------------------
extern "C" void kernel_launch(void* const* d_in, const int* in_sizes, int n_in,
                              void* d_out, int out_size, void* d_ws, size_t ws_size,
                              hipStream_t stream) {
    (void)in_sizes; (void)n_in; (void)out_size; (void)ws_size;
    const float* h   = (const float*)d_in[0];
    const float* uv  = (const float*)d_in[1];
    const float* Kgc = (const float*)d_in[2];
    const float* Kgs = (const float*)d_in[3];
    const float* Kdc = (const float*)d_in[4];
    const float* Kds = (const float*)d_in[5];
    const float* f   = (const float*)d_in[6];
    float* out = (float*)d_out;
    float* ws  = (float*)d_ws;

    float* ch   = ws + 0 * NTOT;
    float* cu   = ws + 1 * NTOT;
    float* cv   = ws + 2 * NTOT;
    float* rawH = ws + 3 * NTOT;
    float* rawC = ws + 4 * NTOT;
    float* rawS = ws + 5 * NTOT;
    float* ah   = ws + 6 * NTOT;
    float* au   = ws + 7 * NTOT;
    float* av   = ws + 8 * NTOT;
    float* KgcT = ws + 9 * NTOT;
    float* KgsT = KgcT + KTSZ;
    float* KdcT = KgsT + KTSZ;
    float* KdsT = KdcT + KTSZ;
    int*   Wt   = (int*)(KdsT + KTSZ);

    setup_w_kernel<<<NTILES, 256, 0, stream>>>(Kgc, Kgs, Kdc, Kds, Wt);
    dim3 tgrid(NLON / TKL, (NLAT + TKL - 1) / TKL);
    transpose_pad_kernel<<<tgrid, 256, 0, stream>>>(Kgc, KgcT);
    transpose_pad_kernel<<<tgrid, 256, 0, stream>>>(Kgs, KgsT);
    transpose_pad_kernel<<<tgrid, 256, 0, stream>>>(Kdc, KdcT);
    transpose_pad_kernel<<<tgrid, 256, 0, stream>>>(Kds, KdsT);

    unsigned nb = (unsigned)((NTOT + 255) / 256);
    init_kernel<<<nb, 256, 0, stream>>>(h, uv, ch, cu, cv, ah, au, av);

    const float waccs[4] = {1.0f, 2.0f, 2.0f, 1.0f};
    const float cs[4]    = {DT_CONST * 0.5f, DT_CONST * 0.5f, DT_CONST, 0.0f};
    dim3 cgrid(NTILES, NB);
    for (int s = 0; s < 4; ++s) {
        // du,dv pre-Coriolis: both gradient kernels share the staged h rows.
        corr_kernel<2, 2><<<cgrid, 256, 0, stream>>>(ch, KgcT, KgsT, rawC, rawS, Wt, 0);
        // dh: corr(Kd_c,u) then += corr(Kd_s,v).
        corr_kernel<1, 4><<<cgrid, 256, 0, stream>>>(cu, KdcT, KdcT, rawH, rawH, Wt, 0);
        corr_kernel<1, 4><<<cgrid, 256, 0, stream>>>(cv, KdsT, KdsT, rawH, rawH, Wt, 1);
        update_kernel<<<nb, 256, 0, stream>>>(h, uv, f, rawH, rawC, rawS,
                                              ch, cu, cv, ah, au, av, out,
                                              waccs[s], cs[s], s == 3 ? 1 : 0);
    }
}